// SelfAttention_31267361915348
// MI455X (gfx1250) — compile-verified
//
#include <hip/hip_runtime.h>
#include <math.h>

#define EPS_YAT (1.0f / 137.0f)
#define BB 4
#define TT 1024
#define CC 768
#define HH 12
#define DD 64
#define F3 2304          // 3*C
#define MM 4096          // B*T
#define NEG_BIG (-3.4028235e38f)

#define KC 32            // K-depth staged per TDM chunk
#define WSTRIDE 80       // LDS row stride in floats (64 data + 16 pad -> bank-conflict free)

typedef float v2f __attribute__((ext_vector_type(2)));
typedef float v8f __attribute__((ext_vector_type(8)));
typedef unsigned int u32x4 __attribute__((ext_vector_type(4)));
typedef int i32x4 __attribute__((ext_vector_type(4)));
typedef int i32x8 __attribute__((ext_vector_type(8)));

static __device__ __forceinline__ v8f wmma4(v2f a, v2f b, v8f c) {
  // D = A(16x4 f32) * B(4x16 f32) + C(16x16 f32)
  return __builtin_amdgcn_wmma_f32_16x16x4_f32(
      /*neg_a=*/false, a, /*neg_b=*/false, b,
      /*c_mod=*/(short)0, c, /*reuse_a=*/false, /*reuse_b=*/false);
}

// LDS byte offset of a __shared__ pointer (generic -> addrspace(3) -> int).
static __device__ __forceinline__ unsigned lds_off(const float* p) {
  return (unsigned)(unsigned long long)(__attribute__((address_space(3))) const float*)p;
}

// Issue one TDM 2D tile load: KC rows x 64 floats from W[k0*N + col0] into LDS,
// with 16-DWORD padding after every 64 DWORDs (row stride WSTRIDE floats).
static __device__ __forceinline__ void tdm_load_w(const float* W, int N, int K,
                                                  int k0, int col0,
                                                  unsigned lds_byte_addr) {
  const unsigned long long gaddr =
      (unsigned long long)(const void*)(W + (size_t)k0 * N + col0);
  u32x4 g0;
  g0[0] = 1u;                                            // count=1, user mode
  g0[1] = lds_byte_addr;                                 // lds_addr [63:32]
  g0[2] = (unsigned)(gaddr & 0xFFFFFFFFu);               // global_addr lo
  g0[3] = (unsigned)((gaddr >> 32) & 0x01FFFFFFu)        // global_addr [56:32]
          | (2u << 30);                                  // type=2 ("image")
  i32x8 g1;
  g1[0] = (int)((2u << 16)        // data_size = 4 bytes
                | (1u << 20)      // pad_enable
                | (5u << 22)      // pad_interval: 64 DWORDs
                | (15u << 25));   // pad_amount: 16 DWORDs
  g1[1] = (int)(((unsigned)N & 0xFFFFu) << 16);          // tensor_dim0[15:0]
  g1[2] = (int)((((unsigned)N >> 16) & 0xFFFFu)          // tensor_dim0[31:16]
                | (((unsigned)K & 0xFFFFu) << 16));      // tensor_dim1[15:0]
  g1[3] = (int)((((unsigned)K >> 16) & 0xFFFFu)          // tensor_dim1[31:16]
                | (64u << 16));                          // tile_dim0 = 64
  g1[4] = (int)(unsigned)KC;                             // tile_dim1 = KC, tile_dim2 = 0
  g1[5] = N;                                             // tensor_dim0_stride lo
  g1[6] = 0;                                             // stride hi, dim1_stride lo
  g1[7] = 0;
  i32x4 zg = {};
#if defined(__clang_major__) && (__clang_major__ >= 23)
  i32x8 z8 = {};
  __builtin_amdgcn_tensor_load_to_lds(g0, g1, zg, zg, z8, 0);
#else
  __builtin_amdgcn_tensor_load_to_lds(g0, g1, zg, zg, 0);
#endif
}

// -------- per-row sum of squares: out[m] = sum_k X[m][k]^2 --------
__global__ __launch_bounds__(256) void row_sumsq(const float* __restrict__ X,
                                                 float* __restrict__ out,
                                                 int Mr, int K) {
  int m = blockIdx.x * blockDim.x + threadIdx.x;
  if (m >= Mr) return;
  const float4* xp = (const float4*)(X + (size_t)m * K);
  float s = 0.0f;
  for (int k = 0; k < (K >> 2); ++k) {
    float4 v = xp[k];
    s += v.x * v.x + v.y * v.y + v.z * v.z + v.w * v.w;
  }
  out[m] = s;
}

// -------- per-column sum of squares: out[n] = sum_k W[k][n]^2 --------
__global__ __launch_bounds__(256) void col_sumsq(const float* __restrict__ W,
                                                 float* __restrict__ out,
                                                 int K, int N) {
  int n = blockIdx.x * blockDim.x + threadIdx.x;
  if (n >= N) return;
  float s = 0.0f;
  for (int k = 0; k < K; ++k) {
    float w = W[(size_t)k * N + n];
    s += w * w;
  }
  out[n] = s;
}

// -------- yat_dense with TDM-staged W panel --------
// Block = 8 waves computes 128 rows x 64 cols. W panel (KC x 64) is DMA'd to
// LDS by the Tensor Data Mover, double-buffered, overlapped with WMMA compute.
__global__ __launch_bounds__(256) void yat_gemm_wmma(
    const float* __restrict__ A, const float* __restrict__ W,
    const float* __restrict__ bias, const float* __restrict__ rown,
    const float* __restrict__ coln, const float* __restrict__ alphap,
    float* __restrict__ out, int Mr, int N, int K) {
  __shared__ float wbuf[2 * KC * WSTRIDE];   // 20 KB double buffer
  const int lane = threadIdx.x & 31;
  const int wv = threadIdx.x >> 5;
  const int ntg = N >> 6;                    // 64-wide column groups
  const int bi = blockIdx.x / ntg;           // 128-row block
  const int bj = blockIdx.x - bi * ntg;
  const int half = lane >> 4;                // 0: K pair {0,1}, 1: {2,3}
  const int lcol = lane & 15;
  const int row = bi * 128 + wv * 16 + lcol; // A-fragment row for this lane
  const int col0 = bj * 64;

  const float* ap = A + (size_t)row * K + 2 * half;
  const unsigned lds0 = lds_off(wbuf);
  const int nch = K / KC;

  if (wv == 0) tdm_load_w(W, N, K, 0, col0, lds0);

  v8f acc0 = {}, acc1 = {}, acc2 = {}, acc3 = {};
  for (int c = 0; c < nch; ++c) {
    const int cur = c & 1;
    if (wv == 0) {
      if (c + 1 < nch) {
        tdm_load_w(W, N, K, (c + 1) * KC, col0,
                   lds0 + (unsigned)((1 - cur) * KC * WSTRIDE * 4));
        __builtin_amdgcn_s_wait_tensorcnt(1);  // chunk c landed; c+1 in flight
      } else {
        __builtin_amdgcn_s_wait_tensorcnt(0);
      }
    }
    __syncthreads();                           // publish buf[cur] to all waves

    const float* lw = wbuf + cur * (KC * WSTRIDE) + 2 * half * WSTRIDE + lcol;
#pragma unroll
    for (int kl = 0; kl < KC; kl += 4) {
      v2f a = *(const v2f*)(ap + c * KC + kl); // A[row][k+2h .. k+2h+1]
      const float* base = lw + kl * WSTRIDE;
      v2f b0, b1, b2, b3;                      // B[k][n] from LDS panel
      b0.x = base[0];  b0.y = base[WSTRIDE];
      b1.x = base[16]; b1.y = base[WSTRIDE + 16];
      b2.x = base[32]; b2.y = base[WSTRIDE + 32];
      b3.x = base[48]; b3.y = base[WSTRIDE + 48];
      acc0 = wmma4(a, b0, acc0);
      acc1 = wmma4(a, b1, acc1);
      acc2 = wmma4(a, b2, acc2);
      acc3 = wmma4(a, b3, acc3);
    }
    __syncthreads();                           // all done reading buf[cur]
  }

  const float alpha = *alphap;
  const float base = sqrtf((float)N) / logf(1.0f + (float)N);
  const float scale = __expf(alpha * __logf(base));

  auto epi = [&](const v8f& av, int t) {
    const int c = col0 + t * 16 + lcol;
    const float wn = coln[c];
    const float bs = bias[c];
#pragma unroll
    for (int e = 0; e < 8; ++e) {
      const int r = bi * 128 + wv * 16 + e + 8 * half;  // C-layout row
      const float dot = av[e];
      const float dist = rown[r] + wn - 2.0f * dot + EPS_YAT;
      out[(size_t)r * N + c] = (dot * dot / dist + bs) * scale;
    }
  };
  epi(acc0, 0); epi(acc1, 1); epi(acc2, 2); epi(acc3, 3);
}

// -------- yat flash attention: one wave per (b,h,16-query tile) --------
__global__ __launch_bounds__(256) void yat_flash_attn(
    const float* __restrict__ qkv, float* __restrict__ aout) {
  __shared__ float ldsbuf[8 * 256];                // 16x16 P tile per wave
  const int lane = threadIdx.x & 31;
  const int wv = threadIdx.x >> 5;
  const int wave = blockIdx.x * 8 + wv;
  const int qt = wave & 63;                        // T/16 = 64 query tiles
  const int h = (wave >> 6) % HH;
  const int b = wave / (64 * HH);
  const int half = lane >> 4;
  const int lcol = lane & 15;

  volatile float* p = ldsbuf + wv * 256;

  const float* qbase = qkv + (size_t)b * TT * F3 + h * DD;
  const float* kbase = qbase + CC;
  const float* vbase = qbase + 2 * CC;

  // Preload Q fragments (16 K-chunks of 4) + full-row q-norm via cross-half shfl.
  const int qrow = qt * 16 + lcol;
  const float* qp = qbase + (size_t)qrow * F3 + 2 * half;
  v2f qf[16];
  float qn = 0.0f;
#pragma unroll
  for (int c = 0; c < 16; ++c) {
    qf[c] = *(const v2f*)(qp + 4 * c);
    qn += qf[c].x * qf[c].x + qf[c].y * qf[c].y;
  }
  qn += __shfl_xor(qn, 16, 32);                    // lane's row (lcol) |q|^2

  v8f o0 = {}, o1 = {}, o2 = {}, o3 = {};
  float rm = NEG_BIG, rl = 0.0f;
  const float rD = 0.125f;                         // 1/sqrt(64)

  for (int j0 = 0; j0 <= qt * 16; j0 += 16) {      // causal: key tiles up to diag
    // S = Q * K^T for this 16x16 tile (16 WMMA over D=64), plus k-norms.
    const float* kp = kbase + (size_t)(j0 + lcol) * F3 + 2 * half;
    float kn = 0.0f;
    v8f s = {};
#pragma unroll
    for (int c = 0; c < 16; ++c) {
      v2f kf = *(const v2f*)(kp + 4 * c);          // B[d][j]: lane=col j, K pair
      kn += kf.x * kf.x + kf.y * kf.y;
      s = wmma4(qf[c], kf, s);
    }
    kn += __shfl_xor(kn, 16, 32);                  // lane's key row (j0+lcol) |k|^2

    const int jcol = j0 + lcol;
#pragma unroll
    for (int e = 0; e < 8; ++e) {
      const int irow = qt * 16 + e + 8 * half;
      const float qne = __shfl(qn, e + 8 * half, 32);
      const float d = s[e] * rD;
      float lg = (d * d) / (qne + kn - 2.0f * d + EPS_YAT);
      if (jcol > irow) lg = NEG_BIG;               // causal mask
      p[(e + 8 * half) * 16 + lcol] = lg;          // C-layout -> row-major LDS
    }
    __builtin_amdgcn_wave_barrier();

    // Online softmax: lane owns row lcol (both halves duplicate, same result).
    float tmax = NEG_BIG;
#pragma unroll
    for (int c2 = 0; c2 < 16; ++c2) tmax = fmaxf(tmax, p[lcol * 16 + c2]);
    const float mnew = fmaxf(rm, tmax);
    const float corr = __expf(rm - mnew);
    float psum = 0.0f;
#pragma unroll
    for (int c2 = 0; c2 < 16; ++c2) {
      const float pe = __expf(p[lcol * 16 + c2] - mnew);
      p[lcol * 16 + c2] = pe;                      // P tile in LDS (A-layout src)
      psum += pe;
    }
    rl = rl * corr + psum;
    rm = mnew;
    __builtin_amdgcn_wave_barrier();

    // Rescale running O by exp(m_old - m_new), per row.
#pragma unroll
    for (int e = 0; e < 8; ++e) {
      const float f = __shfl(corr, e + 8 * half, 32);
      o0[e] *= f; o1[e] *= f; o2[e] *= f; o3[e] *= f;
    }

    // O += P (16x16) x V (16x64): 4 K-chunks x 4 N-tiles = 16 WMMA.
#pragma unroll
    for (int c = 0; c < 4; ++c) {
      v2f pa;                                      // A[row=lcol][4c+2h .. +1]
      pa.x = p[lcol * 16 + 4 * c + 2 * half];
      pa.y = p[lcol * 16 + 4 * c + 2 * half + 1];
      const float* vp0 = vbase + (size_t)(j0 + 4 * c + 2 * half) * F3 + lcol;
      v2f b0, b1, b2, b3;                          // B[k][n]: lane=col n
      b0.x = vp0[0];  b0.y = vp0[F3];
      b1.x = vp0[16]; b1.y = vp0[F3 + 16];
      b2.x = vp0[32]; b2.y = vp0[F3 + 32];
      b3.x = vp0[48]; b3.y = vp0[F3 + 48];
      o0 = wmma4(pa, b0, o0);
      o1 = wmma4(pa, b1, o1);
      o2 = wmma4(pa, b2, o2);
      o3 = wmma4(pa, b3, o3);
    }
  }

  // Finalize: divide by softmax denominator, store [B,T,C] with head offset.
  const float inv = 1.0f / rl;
  float* obase = aout + (size_t)b * TT * CC + h * DD;
#pragma unroll
  for (int e = 0; e < 8; ++e) {
    const float f = __shfl(inv, e + 8 * half, 32);
    const int irow = qt * 16 + e + 8 * half;
    float* orow = obase + (size_t)irow * CC;
    orow[lcol]      = o0[e] * f;
    orow[16 + lcol] = o1[e] * f;
    orow[32 + lcol] = o2[e] * f;
    orow[48 + lcol] = o3[e] * f;
  }
}

extern "C" void kernel_launch(void* const* d_in, const int* in_sizes, int n_in,
                              void* d_out, int out_size, void* d_ws, size_t ws_size,
                              hipStream_t stream) {
  (void)in_sizes; (void)n_in; (void)out_size; (void)ws_size;
  const float* x   = (const float*)d_in[0];
  // d_in[1] = causal mask: implicit in the attention kernel, not read.
  const float* Wa  = (const float*)d_in[2];
  const float* ba  = (const float*)d_in[3];
  const float* ala = (const float*)d_in[4];
  const float* Wp  = (const float*)d_in[5];
  const float* bp  = (const float*)d_in[6];
  const float* alp = (const float*)d_in[7];
  float* out = (float*)d_out;

  float* ws  = (float*)d_ws;
  float* qkv = ws;                                  // [M, 3C]
  float* ao  = qkv + (size_t)MM * F3;               // [M, C] attention output
  float* xn  = ao + (size_t)MM * CC;                // [M] |x|^2
  float* on  = xn + MM;                             // [M] |attn_out|^2
  float* wn1 = on + MM;                             // [3C] |W_attn col|^2
  float* wn2 = wn1 + F3;                            // [C]  |W_proj col|^2

  col_sumsq<<<(F3 + 255) / 256, 256, 0, stream>>>(Wa, wn1, CC, F3);
  col_sumsq<<<(CC + 255) / 256, 256, 0, stream>>>(Wp, wn2, CC, CC);
  row_sumsq<<<MM / 256, 256, 0, stream>>>(x, xn, MM, CC);

  {  // QKV yat_dense: 4096x2304x768, 128x64 tile per block
    int blocks = (MM / 128) * (F3 / 64);
    yat_gemm_wmma<<<blocks, 256, 0, stream>>>(x, Wa, ba, xn, wn1, ala,
                                              qkv, MM, F3, CC);
  }
  {  // Yat attention: B*H*(T/16) waves
    int waves = BB * HH * (TT / 16);
    yat_flash_attn<<<waves / 8, 256, 0, stream>>>(qkv, ao);
  }
  row_sumsq<<<MM / 256, 256, 0, stream>>>(ao, on, MM, CC);
  {  // Projection yat_dense: 4096x768x768
    int blocks = (MM / 128) * (CC / 64);
    yat_gemm_wmma<<<blocks, 256, 0, stream>>>(ao, Wp, bp, on, wn2, alp,
                                              out, MM, CC, CC);
  }
}